// GaussianRasterizer_51273319580251
// MI455X (gfx1250) — compile-verified
//
#include <hip/hip_runtime.h>
#include <math.h>

#define P_N 512
#define IMG_H 256
#define IMG_W 256
#define TANFOVX_C 0.5f
#define TANFOVY_C 0.5f
#define NEAR_C 0.2f
#define COV_BLUR_C 0.3f
#define ALPHA_MIN_C (1.0f/255.0f)
#define ALPHA_MAX_C 0.99f

typedef __attribute__((ext_vector_type(2))) float v2f;
typedef __attribute__((ext_vector_type(8))) float v8f;

// ---------------------------------------------------------------------------
// Kernel 1: per-gaussian preprocess -> depth, power-polynomial coeffs, params
// coef[i][0..7] = {c0, cx, cy, cxx, cxy, cyy, 0, 0}   (power = c · [1,x,y,x²,xy,y²])
// par[i][0..3]  = {opacity (0 if culled), R, G, B}
// ---------------------------------------------------------------------------
__global__ __launch_bounds__(256) void gs_preprocess(
    const float* __restrict__ means3D, const float* __restrict__ opac,
    const float* __restrict__ scales,  const float* __restrict__ rots,
    const float* __restrict__ cols,    const float* __restrict__ vm,
    const float* __restrict__ pm,
    float* __restrict__ depth, float* __restrict__ coef, float* __restrict__ par)
{
    int i = blockIdx.x * blockDim.x + threadIdx.x;
    if (i >= P_N) return;

    float mx = means3D[3*i+0], my = means3D[3*i+1], mz = means3D[3*i+2];

    // view transform t = V[:3,:3] @ m + V[:3,3]
    float tx = vm[0]*mx + vm[1]*my + vm[2]*mz  + vm[3];
    float ty = vm[4]*mx + vm[5]*my + vm[6]*mz  + vm[7];
    float tz = vm[8]*mx + vm[9]*my + vm[10]*mz + vm[11];
    bool valid = tz > NEAR_C;
    float tzc = valid ? tz : 1.0f;

    // quaternion -> rotation
    float qr = rots[4*i+0], qx = rots[4*i+1], qy = rots[4*i+2], qz = rots[4*i+3];
    float qn = sqrtf(qr*qr + qx*qx + qy*qy + qz*qz);
    qn = fmaxf(qn, 1e-12f);
    qr /= qn; qx /= qn; qy /= qn; qz /= qn;
    float R00 = 1.f - 2.f*(qy*qy + qz*qz), R01 = 2.f*(qx*qy - qr*qz), R02 = 2.f*(qx*qz + qr*qy);
    float R10 = 2.f*(qx*qy + qr*qz), R11 = 1.f - 2.f*(qx*qx + qz*qz), R12 = 2.f*(qy*qz - qr*qx);
    float R20 = 2.f*(qx*qz - qr*qy), R21 = 2.f*(qy*qz + qr*qx), R22 = 1.f - 2.f*(qx*qx + qy*qy);

    float s0 = expf(scales[3*i+0]), s1 = expf(scales[3*i+1]), s2 = expf(scales[3*i+2]);
    float L00 = R00*s0, L01 = R01*s1, L02 = R02*s2;
    float L10 = R10*s0, L11 = R11*s1, L12 = R12*s2;
    float L20 = R20*s0, L21 = R21*s1, L22 = R22*s2;

    // cov3D = L L^T (symmetric)
    float C00 = L00*L00 + L01*L01 + L02*L02;
    float C01 = L00*L10 + L01*L11 + L02*L12;
    float C02 = L00*L20 + L01*L21 + L02*L22;
    float C11 = L10*L10 + L11*L11 + L12*L12;
    float C12 = L10*L20 + L11*L21 + L12*L22;
    float C22 = L20*L20 + L21*L21 + L22*L22;

    const float fx = (float)IMG_W / (2.0f * TANFOVX_C);
    const float fy = (float)IMG_H / (2.0f * TANFOVY_C);
    const float limx = 1.3f * TANFOVX_C, limy = 1.3f * TANFOVY_C;
    float invz = 1.0f / tzc;
    float txz = fminf(fmaxf(tx*invz, -limx), limx) * tzc;
    float tyz = fminf(fmaxf(ty*invz, -limy), limy) * tzc;
    float J00 = fx*invz, J02 = -fx*txz*invz*invz;
    float J11 = fy*invz, J12 = -fy*tyz*invz*invz;

    // T = J @ V[:3,:3]
    float T00 = J00*vm[0] + J02*vm[8];
    float T01 = J00*vm[1] + J02*vm[9];
    float T02 = J00*vm[2] + J02*vm[10];
    float T10 = J11*vm[4] + J12*vm[8];
    float T11 = J11*vm[5] + J12*vm[9];
    float T12 = J11*vm[6] + J12*vm[10];

    // cov2D = T cov3D T^T + blur*I
    float M00 = T00*C00 + T01*C01 + T02*C02;
    float M01 = T00*C01 + T01*C11 + T02*C12;
    float M02 = T00*C02 + T01*C12 + T02*C22;
    float M10 = T10*C00 + T11*C01 + T12*C02;
    float M11 = T10*C01 + T11*C11 + T12*C12;
    float M12 = T10*C02 + T11*C12 + T12*C22;
    float A  = M00*T00 + M01*T01 + M02*T02 + COV_BLUR_C;
    float Bv = M00*T10 + M01*T11 + M02*T12;
    float Cv = M10*T10 + M11*T11 + M12*T12 + COV_BLUR_C;
    float det = A*Cv - Bv*Bv;
    det = (det == 0.0f) ? 1.0f : det;
    float conA =  Cv / det;
    float conB = -Bv / det;
    float conC =  A  / det;

    // projection
    float ph0 = pm[0]*mx  + pm[1]*my  + pm[2]*mz  + pm[3];
    float ph1 = pm[4]*mx  + pm[5]*my  + pm[6]*mz  + pm[7];
    float ph3 = pm[12]*mx + pm[13]*my + pm[14]*mz + pm[15];
    float pw = 1.0f / (ph3 + 1e-7f);
    float px = ((ph0*pw + 1.0f) * (float)IMG_W - 1.0f) * 0.5f;
    float py = ((ph1*pw + 1.0f) * (float)IMG_H - 1.0f) * 0.5f;

    float op = 1.0f / (1.0f + expf(-opac[i]));
    if (!valid) op = 0.0f;   // forces alpha=0 < ALPHA_MIN -> zeroed, matching ~valid

    // power(x,y) = c0 + cx*x + cy*y + cxx*x^2 + cxy*x*y + cyy*y^2
    float c0  = -0.5f*conA*px*px - conB*px*py - 0.5f*conC*py*py;
    float cx  =  conA*px + conB*py;
    float cy  =  conB*px + conC*py;
    float cxx = -0.5f*conA;
    float cxy = -conB;
    float cyy = -0.5f*conC;

    float* cf = coef + i*8;
    cf[0] = c0; cf[1] = cx; cf[2] = cy; cf[3] = cxx;
    cf[4] = cxy; cf[5] = cyy; cf[6] = 0.0f; cf[7] = 0.0f;
    float* pp = par + i*4;
    pp[0] = op; pp[1] = cols[3*i+0]; pp[2] = cols[3*i+1]; pp[3] = cols[3*i+2];
    depth[i] = valid ? tz : INFINITY;
}

// ---------------------------------------------------------------------------
// Kernel 2: stable rank sort by depth (ties broken by index == stable argsort)
// ---------------------------------------------------------------------------
__global__ __launch_bounds__(P_N) void gs_sort(
    const float* __restrict__ depth, const float* __restrict__ coef,
    const float* __restrict__ par, float* __restrict__ scoef, float* __restrict__ spar)
{
    __shared__ float sd[P_N];
    int i = threadIdx.x;
    sd[i] = depth[i];
    __syncthreads();
    float di = sd[i];
    int rank = 0;
    for (int j = 0; j < P_N; ++j) {
        float dj = sd[j];
        rank += (dj < di) || (dj == di && j < i);
    }
#pragma unroll
    for (int k = 0; k < 8; ++k) scoef[rank*8 + k] = coef[i*8 + k];
#pragma unroll
    for (int k = 0; k < 4; ++k) spar[rank*4 + k] = par[i*4 + k];
}

// ---------------------------------------------------------------------------
// Kernel 3: rasterize. One wave owns 16 pixels; per 16-gaussian chunk the
// 16x16 power tile comes from two V_WMMA_F32_16X16X4_F32 (rank-6 quadratic
// form as a K=4+K=4 GEMM); blend is segmented front-to-back with a
// shfl_xor(16) combine across the two half-waves (M=0..7 in lanes 0-15,
// M=8..15 in lanes 16-31). The sorted gaussian table (24 KB) is staged into
// LDS once per block via async global->LDS (ASYNCcnt) loads.
// ---------------------------------------------------------------------------
__global__ __launch_bounds__(256) void gs_raster(
    const float* __restrict__ scoef, const float* __restrict__ spar,
    const float* __restrict__ bg, float* __restrict__ out)
{
    __shared__ __align__(16) float s_coef[P_N * 8];  // 16 KB
    __shared__ __align__(16) float s_par[P_N * 4];   //  8 KB

    int t = threadIdx.x;

    // ---- async stage global -> LDS (CDNA5 async path, tracked by ASYNCcnt) ----
    {
        unsigned lc = (unsigned)(uintptr_t)s_coef;   // workgroup-relative LDS offset
        unsigned lp = (unsigned)(uintptr_t)s_par;
#pragma unroll
        for (int p = 0; p < 4; ++p) {                // 4 * 4096B = 16 KB coeffs
            unsigned loff = lc + p * 4096 + t * 16;
            const char* ga = (const char*)scoef + p * 4096 + t * 16;
            asm volatile("global_load_async_to_lds_b128 %0, %1, off"
                         :: "v"(loff), "v"(ga) : "memory");
        }
#pragma unroll
        for (int p = 0; p < 2; ++p) {                // 2 * 4096B = 8 KB params
            unsigned loff = lp + p * 4096 + t * 16;
            const char* ga = (const char*)spar + p * 4096 + t * 16;
            asm volatile("global_load_async_to_lds_b128 %0, %1, off"
                         :: "v"(loff), "v"(ga) : "memory");
        }
        asm volatile("s_wait_asynccnt 0x0" ::: "memory");
        __syncthreads();
    }

    int tid  = blockIdx.x * blockDim.x + t;
    int lane = t & 31;
    int wave = tid >> 5;
    int sub  = lane & 15;     // pixel column within tile / gaussian row
    int hi   = lane >> 4;     // half-wave selector

    int   pid = wave * 16 + sub;          // 0..65535, exact coverage
    float x = (float)(pid & (IMG_W - 1));
    float y = (float)(pid >> 8);

    // B fragment (4x16): VGPR e holds row K = e + 2*hi
    float m0 = 1.0f, m1 = x, m2 = y, m3 = x*x, m4 = x*y, m5 = y*y;
    v2f b1, b2;
    b1.x = hi ? m2 : m0;      // K0=1   | K2=y
    b1.y = hi ? m3 : m1;      // K1=x   | K3=x^2
    b2.x = hi ? 0.0f : m4;    // K4=x*y | K6=0
    b2.y = hi ? 0.0f : m5;    // K5=y^2 | K7=0

    float accR = 0.0f, accG = 0.0f, accB = 0.0f, Trun = 1.0f;

    for (int chunk = 0; chunk < P_N / 16; ++chunk) {
        // A fragment (16x4): lane row M=sub, VGPR e holds K = e + 2*hi
        int gA = chunk * 16 + sub;
        const float* cp = s_coef + gA * 8 + 2 * hi;
        v2f a1, a2;
        a1.x = cp[0]; a1.y = cp[1];   // c[2h], c[2h+1]
        a2.x = cp[4]; a2.y = cp[5];   // c[4+2h], c[5+2h] (zeros for hi=1)

        v8f pw8 = {0.f,0.f,0.f,0.f,0.f,0.f,0.f,0.f};
        pw8 = __builtin_amdgcn_wmma_f32_16x16x4_f32(false, a1, false, b1, (short)0, pw8, false, false);
        pw8 = __builtin_amdgcn_wmma_f32_16x16x4_f32(false, a2, false, b2, (short)0, pw8, false, false);

        // local blend of this half's 8 gaussians (M = 8*hi + r), front-to-back
        float cr = 0.f, cg = 0.f, cb = 0.f, Tc = 1.f;
#pragma unroll
        for (int r = 0; r < 8; ++r) {
            int g = chunk * 16 + hi * 8 + r;
            float4 p = ((const float4*)s_par)[g];     // ds_load_b128
            float power = pw8[r];
            float alpha = fminf(ALPHA_MAX_C, p.x * __expf(power));
            alpha = (power > 0.0f || alpha < ALPHA_MIN_C) ? 0.0f : alpha;
            float w = Tc * alpha;
            cr += w * p.y; cg += w * p.z; cb += w * p.w;
            Tc *= (1.0f - alpha);
        }

        // combine the two ordered 8-segments across half-waves
        float orr = __shfl_xor(cr, 16, 32);
        float org = __shfl_xor(cg, 16, 32);
        float orb = __shfl_xor(cb, 16, 32);
        float orT = __shfl_xor(Tc, 16, 32);
        float fr = hi ? orr : cr,  fg = hi ? org : cg,  fb = hi ? orb : cb,  fT = hi ? orT : Tc;
        float br = hi ? cr : orr,  bgc = hi ? cg : org, bb = hi ? cb : orb;
        float ccr = fr + fT * br;
        float ccg = fg + fT * bgc;
        float ccb = fb + fT * bb;
        float ccT = Tc * orT;

        accR += Trun * ccr;
        accG += Trun * ccg;
        accB += Trun * ccb;
        Trun *= ccT;
    }

    if (hi == 0) {
        out[pid*3 + 0] = accR + Trun * bg[0];
        out[pid*3 + 1] = accG + Trun * bg[1];
        out[pid*3 + 2] = accB + Trun * bg[2];
    }
}

// ---------------------------------------------------------------------------
extern "C" void kernel_launch(void* const* d_in, const int* in_sizes, int n_in,
                              void* d_out, int out_size, void* d_ws, size_t ws_size,
                              hipStream_t stream)
{
    const float* means3D = (const float*)d_in[0];
    const float* opac    = (const float*)d_in[1];
    const float* scales  = (const float*)d_in[2];
    const float* rots    = (const float*)d_in[3];
    const float* cols    = (const float*)d_in[4];
    const float* vm      = (const float*)d_in[5];
    const float* pm      = (const float*)d_in[6];
    const float* bg      = (const float*)d_in[8];

    char* ws = (char*)d_ws;
    float* depth = (float*)(ws);                         //  512 f
    float* coef  = (float*)(ws + 2048);                  //  512*8 f
    float* par   = (float*)(ws + 2048 + 16384);          //  512*4 f
    float* scoef = (float*)(ws + 2048 + 16384 + 8192);   //  512*8 f
    float* spar  = (float*)(ws + 2048 + 16384 + 8192 + 16384); // 512*4 f

    gs_preprocess<<<P_N/256, 256, 0, stream>>>(means3D, opac, scales, rots, cols,
                                               vm, pm, depth, coef, par);
    gs_sort<<<1, P_N, 0, stream>>>(depth, coef, par, scoef, spar);
    gs_raster<<<(IMG_H*IMG_W/16)/8, 256, 0, stream>>>(scoef, spar, bg, (float*)d_out);
}